// GCNs_21260088115544
// MI455X (gfx1250) — compile-verified
//
#include <hip/hip_runtime.h>

// Problem constants (match reference)
#define N_NODES 50000
#define N_EDGES 800000
#define D_IN    128
#define HID     128
#define D_OUT   64

typedef __attribute__((ext_vector_type(2))) float v2f;
typedef __attribute__((ext_vector_type(4))) float v4f;
typedef __attribute__((ext_vector_type(8))) float v8f;

// ---------------------------------------------------------------------------
// Scatter-add aggregation: Agg[dst[e]][:] += X[src[e]][:], rows of 128 floats.
// One wave32 per edge: each lane loads float4 (coalesced 512B row) and issues
// 4 hardware global_atomic_add_f32 (non-returning -> STOREcnt path).
// ---------------------------------------------------------------------------
__global__ void __launch_bounds__(256)
scatter_add_128(const float* __restrict__ X,
                const int*   __restrict__ src,
                const int*   __restrict__ dst,
                float*       __restrict__ Agg,
                int nEdges)
{
    const int wavesPerBlock = blockDim.x >> 5;
    const int e = blockIdx.x * wavesPerBlock + (threadIdx.x >> 5);
    if (e >= nEdges) return;

    const int lane = threadIdx.x & 31;
    const int s = src[e];
    const int d = dst[e];

    const v4f v = *(const v4f*)(X + (size_t)s * 128 + lane * 4);
    float* p = Agg + (size_t)d * 128 + lane * 4;

    (void)__hip_atomic_fetch_add(p + 0, v.x, __ATOMIC_RELAXED, __HIP_MEMORY_SCOPE_AGENT);
    (void)__hip_atomic_fetch_add(p + 1, v.y, __ATOMIC_RELAXED, __HIP_MEMORY_SCOPE_AGENT);
    (void)__hip_atomic_fetch_add(p + 2, v.z, __ATOMIC_RELAXED, __HIP_MEMORY_SCOPE_AGENT);
    (void)__hip_atomic_fetch_add(p + 3, v.w, __ATOMIC_RELAXED, __HIP_MEMORY_SCOPE_AGENT);
}

// ---------------------------------------------------------------------------
// Fused GraphConv GEMM: Out = [relu](Agg * Wrel^T + X * Wroot^T + bias)
//   Agg, X : [N, DINT] row-major
//   Wrel, Wroot : [DOUTT, DINT] row-major (PyG Linear layout, y = x W^T)
// One wave per 16x16 output tile; block = one 16-row strip, DOUTT/16 waves.
// Uses V_WMMA_F32_16X16X4_F32 (fp32 matrix cores, full precision).
//
// A-matrix (16x4 f32) lane layout: M = lane&15, K = (lane>>4)*2 + {0,1}
// B-matrix (4x16 f32) lane layout: N = lane&15, K = (lane>>4)*2 + {0,1}
// C/D (16x16 f32): VGPR i -> row M = i + (lane>>4)*8, col N = lane&15
// ---------------------------------------------------------------------------
template<int DINT, int DOUTT, bool RELU>
__global__ void __launch_bounds__((DOUTT / 16) * 32)
graphconv_gemm(const float* __restrict__ Agg,
               const float* __restrict__ X,
               const float* __restrict__ Wrel,
               const float* __restrict__ Wroot,
               const float* __restrict__ bias,
               float*       __restrict__ Out)
{
    const int lane = threadIdx.x & 31;
    const int half = lane >> 4;     // 0: lanes 0-15, 1: lanes 16-31
    const int l15  = lane & 15;
    const int tileN = threadIdx.x >> 5;     // which 16-col tile of output
    const int rowBase = blockIdx.x * 16;    // 16-row strip

    const float* aRelP  = Agg   + (size_t)(rowBase + l15) * DINT + half * 2;
    const float* aRootP = X     + (size_t)(rowBase + l15) * DINT + half * 2;
    const float* bRelP  = Wrel  + (size_t)(tileN * 16 + l15) * DINT + half * 2;
    const float* bRootP = Wroot + (size_t)(tileN * 16 + l15) * DINT + half * 2;

    v8f accRel  = {};
    v8f accRoot = {};

#pragma unroll
    for (int k = 0; k < DINT; k += 4) {
        v2f aRel  = *(const v2f*)(aRelP  + k);
        v2f bRel  = *(const v2f*)(bRelP  + k);
        v2f aRoot = *(const v2f*)(aRootP + k);
        v2f bRoot = *(const v2f*)(bRootP + k);
        // Two independent accumulation chains (accumulate via C: no D->A/B
        // RAW hazard between consecutive WMMAs).
        accRel  = __builtin_amdgcn_wmma_f32_16x16x4_f32(
                      false, aRel,  false, bRel,  (short)0, accRel,  false, false);
        accRoot = __builtin_amdgcn_wmma_f32_16x16x4_f32(
                      false, aRoot, false, bRoot, (short)0, accRoot, false, false);
    }

    const float bb = bias[tileN * 16 + l15];
    float* outp = Out + (size_t)(rowBase + half * 8) * DOUTT + tileN * 16 + l15;

#pragma unroll
    for (int i = 0; i < 8; ++i) {
        float v = accRel[i] + accRoot[i] + bb;
        if (RELU) v = v > 0.0f ? v : 0.0f;
        outp[(size_t)i * DOUTT] = v;
    }
}

// ---------------------------------------------------------------------------
// Launcher. Inputs (setup_inputs order):
//   0: x [50000,128] f32       1: edge_index [2,800000] int
//   2: W_rel1 [128,128] f32    3: W_root1 [128,128] f32   4: b1 [128] f32
//   5: W_rel2 [64,128]  f32    6: W_root2 [64,128]  f32   7: b2 [64]  f32
// Output: [50000, 64] f32
// Workspace: agg [50000*128] f32 (reused both layers) + h [50000*128] f32
// ---------------------------------------------------------------------------
extern "C" void kernel_launch(void* const* d_in, const int* in_sizes, int n_in,
                              void* d_out, int out_size, void* d_ws, size_t ws_size,
                              hipStream_t stream)
{
    const float* x      = (const float*)d_in[0];
    const int*   ei     = (const int*)  d_in[1];
    const float* Wrel1  = (const float*)d_in[2];
    const float* Wroot1 = (const float*)d_in[3];
    const float* b1     = (const float*)d_in[4];
    const float* Wrel2  = (const float*)d_in[5];
    const float* Wroot2 = (const float*)d_in[6];
    const float* b2     = (const float*)d_in[7];
    float* out = (float*)d_out;

    float* agg = (float*)d_ws;                      // [N_NODES * 128]
    float* h   = agg + (size_t)N_NODES * HID;       // [N_NODES * 128]

    const int* src = ei;             // edge_index[0]
    const int* dst = ei + N_EDGES;   // edge_index[1]

    const size_t aggBytes = (size_t)N_NODES * HID * sizeof(float);
    const int scatterBlocks = (N_EDGES + 7) / 8;    // 8 waves (edges) / block

    // ---- Layer 1 ----
    hipMemsetAsync(agg, 0, aggBytes, stream);
    scatter_add_128<<<scatterBlocks, 256, 0, stream>>>(x, src, dst, agg, N_EDGES);
    graphconv_gemm<D_IN, HID, true>
        <<<N_NODES / 16, (HID / 16) * 32, 0, stream>>>(agg, x, Wrel1, Wroot1, b1, h);

    // ---- Layer 2 ----
    hipMemsetAsync(agg, 0, aggBytes, stream);
    scatter_add_128<<<scatterBlocks, 256, 0, stream>>>(h, src, dst, agg, N_EDGES);
    graphconv_gemm<HID, D_OUT, false>
        <<<N_NODES / 16, (D_OUT / 16) * 32, 0, stream>>>(agg, h, Wrel2, Wroot2, b2, out);
}